// AttentionPooling_29686813950821
// MI455X (gfx1250) — compile-verified
//
#include <hip/hip_runtime.h>
#include <hip/hip_bf16.h>

#define D 128
#define H 64
#define NSEG 1024
#define POOL_ROWS 256

typedef float v2f __attribute__((ext_vector_type(2)));
typedef float v4f __attribute__((ext_vector_type(4)));
typedef float v8f __attribute__((ext_vector_type(8)));

__device__ __forceinline__ void atomicMaxF32(float* addr, float val) {
    // sign-magnitude trick: correct total order for IEEE floats incl. negatives
    if (val >= 0.0f) {
        atomicMax((int*)addr, __float_as_int(val));
    } else {
        atomicMin((unsigned int*)addr, __float_as_uint(val));
    }
}

// ---------------------------------------------------------------- init
__global__ void ap_init_kernel(float* __restrict__ smax, float* __restrict__ denom,
                               float* __restrict__ counts, float* __restrict__ sums) {
    int i = blockIdx.x * blockDim.x + threadIdx.x;
    if (i < NSEG) {
        smax[i]   = -__builtin_inff();
        denom[i]  = 0.0f;
        counts[i] = 0.0f;
    }
    if (i < NSEG * D) sums[i] = 0.0f;
}

// ---------------------------------------------------------------- scores (WMMA)
// One wave32 per 16-row tile. S_tile[16x64] = X_tile[16x128] @ W1[128x64],
// accumulated with V_WMMA_F32_16X16X4_F32 (4 N-tiles x 32 K-steps).
__global__ void ap_scores_kernel(const float* __restrict__ x,
                                 const int*   __restrict__ batch,
                                 const float* __restrict__ W1,
                                 const float* __restrict__ b1,
                                 const float* __restrict__ W2,
                                 const float* __restrict__ b2,
                                 float* __restrict__ s,
                                 float* __restrict__ smax,
                                 int nTiles, int N) {
    const int wave = threadIdx.x >> 5;
    const int lane = threadIdx.x & 31;
    const int tile = blockIdx.x * (blockDim.x >> 5) + wave;
    if (tile >= nTiles) return;              // wave-uniform: EXEC stays all-ones

    const int r0   = tile * 16;
    const int m    = lane & 15;              // M index (A) / N index (B,C,D)
    const int hi   = lane >> 4;              // lane half
    const int koff = hi * 2;                 // K sub-offset for A/B fragments

    // clamp row for loads on a (never-occurring with N%16==0) partial tile
    int rowA = r0 + m; if (rowA > N - 1) rowA = N - 1;
    const float* xrow = x + (size_t)rowA * D;

    v8f acc0 = {}, acc1 = {}, acc2 = {}, acc3 = {};

#pragma unroll 4
    for (int k0 = 0; k0 < D; k0 += 4) {
        v2f a;
        a.x = xrow[k0 + koff];
        a.y = xrow[k0 + koff + 1];
        const float* w1k0 = W1 + (size_t)(k0 + koff) * H;
        const float* w1k1 = W1 + (size_t)(k0 + koff + 1) * H;
        v2f b;
        b.x = w1k0[0 * 16 + m]; b.y = w1k1[0 * 16 + m];
        acc0 = __builtin_amdgcn_wmma_f32_16x16x4_f32(false, a, false, b, (short)0, acc0, false, false);
        b.x = w1k0[1 * 16 + m]; b.y = w1k1[1 * 16 + m];
        acc1 = __builtin_amdgcn_wmma_f32_16x16x4_f32(false, a, false, b, (short)0, acc1, false, false);
        b.x = w1k0[2 * 16 + m]; b.y = w1k1[2 * 16 + m];
        acc2 = __builtin_amdgcn_wmma_f32_16x16x4_f32(false, a, false, b, (short)0, acc2, false, false);
        b.x = w1k0[3 * 16 + m]; b.y = w1k1[3 * 16 + m];
        acc3 = __builtin_amdgcn_wmma_f32_16x16x4_f32(false, a, false, b, (short)0, acc3, false, false);
    }

    // epilogue: +b1, tanh, dot with W2 over 64 columns
    float b1v0 = b1[0 * 16 + m], b1v1 = b1[1 * 16 + m];
    float b1v2 = b1[2 * 16 + m], b1v3 = b1[3 * 16 + m];
    float w2v0 = W2[0 * 16 + m], w2v1 = W2[1 * 16 + m];
    float w2v2 = W2[2 * 16 + m], w2v3 = W2[3 * 16 + m];
    float bias2 = b2[0];

    float partial[8];
#pragma unroll
    for (int j = 0; j < 8; ++j) {
        float p;
        p  = tanhf(acc0[j] + b1v0) * w2v0;
        p += tanhf(acc1[j] + b1v1) * w2v1;
        p += tanhf(acc2[j] + b1v2) * w2v2;
        p += tanhf(acc3[j] + b1v3) * w2v3;
        partial[j] = p;
    }
    // reduce across the 16 lanes of each half (masks 8..1 stay within a half).
    // C/D layout: VGPR j -> row j (lanes 0-15) / row j+8 (lanes 16-31).
#pragma unroll
    for (int j = 0; j < 8; ++j) {
#pragma unroll
        for (int msk = 8; msk >= 1; msk >>= 1)
            partial[j] += __shfl_xor(partial[j], msk, 32);
    }

    if (m == 0) {
#pragma unroll
        for (int j = 0; j < 8; ++j) {
            int row = r0 + j + hi * 8;
            if (row < N) {
                float sv = partial[j] + bias2;
                s[row] = sv;
                atomicMaxF32(&smax[batch[row]], sv);
            }
        }
    }
}

// ---------------------------------------------------------------- exp + denom
// Wave-aggregated atomics: batch is sorted, so a wave's 32 ids are almost
// always uniform -> one atomicAdd per wave instead of 32.
__global__ void ap_exp_kernel(const int* __restrict__ batch,
                              const float* __restrict__ smax,
                              float* __restrict__ s_inout,   // s in, e out (same buffer)
                              float* __restrict__ denom,
                              float* __restrict__ counts, int N) {
    const int i = blockIdx.x * blockDim.x + threadIdx.x;
    const int lane = threadIdx.x & 31;
    // wave-uniform: is every lane of this wave in range?
    const bool fullWave = (blockIdx.x * blockDim.x + (threadIdx.x | 31)) < N;
    if (i >= N) return;

    int b = batch[i];
    float e = expf(s_inout[i] - smax[b]);
    s_inout[i] = e;

    if (fullWave) {
        int b0 = __builtin_amdgcn_readfirstlane(b);
        bool uni = __all(b == b0);             // wave-uniform result
        if (uni) {
            float tot = e;
#pragma unroll
            for (int msk = 16; msk >= 1; msk >>= 1)
                tot += __shfl_xor(tot, msk, 32);
            if (lane == 0) {
                atomicAdd(&denom[b0], tot);
                atomicAdd(&counts[b0], 32.0f);
            }
            return;
        }
    }
    atomicAdd(&denom[b], e);
    atomicAdd(&counts[b], 1.0f);
}

// ---------------------------------------------------------------- weighted pool
// 4 waves per block, one row per wave per step (32 lanes x float4 = 128 dims,
// one global_load_b128 nt per row). Local accumulation across the sorted rows;
// flush to global atomics only at segment boundaries.
__global__ void ap_pool_kernel(const float* __restrict__ x,
                               const int*   __restrict__ batch,
                               const float* __restrict__ e,
                               const float* __restrict__ denom,
                               float* __restrict__ sums, int N) {
    const int wave = threadIdx.x >> 5;              // 0..3
    const int lane = threadIdx.x & 31;
    const int d4   = lane * 4;                      // this lane's 4 dims
    const int base = blockIdx.x * POOL_ROWS;
    const int row0 = base + wave;
    if (row0 >= N) return;
    int rowEnd = base + POOL_ROWS;
    if (rowEnd > N) rowEnd = N;

    v4f acc = {0.0f, 0.0f, 0.0f, 0.0f};
    int curSeg = batch[row0];
    for (int r = row0; r < rowEnd; r += 4) {
        int b = batch[r];
        if (b != curSeg) {
            float* dst = &sums[(size_t)curSeg * D + d4];
            atomicAdd(dst + 0, acc.x);
            atomicAdd(dst + 1, acc.y);
            atomicAdd(dst + 2, acc.z);
            atomicAdd(dst + 3, acc.w);
            acc = (v4f){0.0f, 0.0f, 0.0f, 0.0f};
            curSeg = b;
        }
        float w = e[r] / denom[b];
        v4f xv = __builtin_nontemporal_load((const v4f*)(x + (size_t)r * D) + lane);
        acc += w * xv;
    }
    float* dst = &sums[(size_t)curSeg * D + d4];
    atomicAdd(dst + 0, acc.x);
    atomicAdd(dst + 1, acc.y);
    atomicAdd(dst + 2, acc.z);
    atomicAdd(dst + 3, acc.w);
}

// ---------------------------------------------------------------- finalize
__global__ void ap_finalize_kernel(const float* __restrict__ sums,
                                   const float* __restrict__ counts,
                                   float* __restrict__ out) {
    int i = blockIdx.x * blockDim.x + threadIdx.x;
    if (i < NSEG * D) out[i] = sums[i] / fmaxf(counts[i >> 7], 1.0f);
}

extern "C" void kernel_launch(void* const* d_in, const int* in_sizes, int n_in,
                              void* d_out, int out_size, void* d_ws, size_t ws_size,
                              hipStream_t stream) {
    const float* x     = (const float*)d_in[0];
    const int*   batch = (const int*)  d_in[1];
    const float* W1    = (const float*)d_in[2];
    const float* b1    = (const float*)d_in[3];
    const float* W2    = (const float*)d_in[4];
    const float* b2    = (const float*)d_in[5];
    const int N = in_sizes[1];

    float* ws     = (float*)d_ws;
    float* s      = ws;              // N floats (scores, then exp weights)
    float* smax   = s + N;           // NSEG
    float* denom  = smax + NSEG;     // NSEG
    float* counts = denom + NSEG;    // NSEG
    float* sums   = counts + NSEG;   // NSEG*D
    float* out    = (float*)d_out;   // NSEG*D

    ap_init_kernel<<<(NSEG * D + 255) / 256, 256, 0, stream>>>(smax, denom, counts, sums);

    const int nTiles = (N + 15) / 16;
    const int wavesPerBlock = 8;     // 256 threads = 8 wave32s
    const int scoreBlocks = (nTiles + wavesPerBlock - 1) / wavesPerBlock;
    ap_scores_kernel<<<scoreBlocks, 256, 0, stream>>>(x, batch, W1, b1, W2, b2,
                                                      s, smax, nTiles, N);

    ap_exp_kernel<<<(N + 255) / 256, 256, 0, stream>>>(batch, smax, s, denom, counts, N);

    ap_pool_kernel<<<(N + POOL_ROWS - 1) / POOL_ROWS, 128, 0, stream>>>(x, batch, s, denom, sums, N);

    ap_finalize_kernel<<<(NSEG * D + 255) / 256, 256, 0, stream>>>(sums, counts, out);
}